// RoIPooling_74122545594381
// MI455X (gfx1250) — compile-verified
//
#include <hip/hip_runtime.h>
#include <stdint.h>

// RoI-align (crop_and_resize bilinear 7x7) for fm [8,64,64,256] f32, boxes [8,1000,4].
// Memory-bound: 401 MB out + 33.5 MB in (L2-resident) -> ~19us floor @ 23.3 TB/s.
// CDNA5 path: async global->LDS staging (ASYNCcnt) double-buffered per wave,
// s_wait_asynccnt handoff, ds_load_b128 reads, nontemporal b128 stores.

typedef __attribute__((ext_vector_type(4))) float v4f;

#define WAVES 4
#define TOTAL_WORK (8 * 1000 * 7)   // one wave per (roi, iy) row

__global__ __launch_bounds__(WAVES * 32)
void roi_align_kernel(const float* __restrict__ fm,
                      const float* __restrict__ boxes,
                      float* __restrict__ out)
{
    // [wave][buf][corner][channel] : 4*2*4*256*4B = 32 KB
    __shared__ __align__(16) float lds[WAVES][2][4][256];

    const int lane = threadIdx.x & 31;
    const int wave = threadIdx.x >> 5;
    const int work = blockIdx.x * WAVES + wave;   // wave-uniform
    if (work >= TOTAL_WORK) return;

    const int r  = work / 7;        // roi id 0..7999
    const int iy = work - r * 7;    // 0..6
    const int b  = r / 1000;        // batch

    const float y1b = boxes[r * 4 + 0];
    const float x1b = boxes[r * 4 + 1];
    const float y2b = boxes[r * 4 + 2];
    const float x2b = boxes[r * 4 + 3];

    // match reference association: (y2-y1)*(H-1)/(crop_h-1)
    const float stepy = (y2b - y1b) * 63.0f / 6.0f;
    const float stepx = (x2b - x1b) * 63.0f / 6.0f;

    const float y   = y1b * 63.0f + (float)iy * stepy;
    const float y0f = floorf(y);
    const float wy  = y - y0f;
    int y0 = (int)y0f; y0 = y0 < 0 ? 0 : (y0 > 63 ? 63 : y0);
    int y1i = y0 + 1;  y1i = y1i > 63 ? 63 : y1i;
    const bool iny = (y >= 0.0f) && (y <= 63.0f);

    const size_t row0 = (size_t)(b * 64 + y0)  * 64 * 256;
    const size_t row1 = (size_t)(b * 64 + y1i) * 64 * 256;

    // LDS byte offset of this wave's staging area (low 32 bits of the flat
    // address of a __shared__ object == its LDS offset on gfx1250).
    const unsigned ldsW = (unsigned)(size_t)&lds[wave][0][0][0];

    auto stage = [&](int ix, int bufSel) {
        const float x = x1b * 63.0f + (float)ix * stepx;
        int x0 = (int)floorf(x); x0 = x0 < 0 ? 0 : (x0 > 63 ? 63 : x0);
        int x1c = x0 + 1;        x1c = x1c > 63 ? 63 : x1c;
        const size_t cols[4] = { row0 + (size_t)x0  * 256,
                                 row0 + (size_t)x1c * 256,
                                 row1 + (size_t)x0  * 256,
                                 row1 + (size_t)x1c * 256 };
        const unsigned bufBase = ldsW + (unsigned)bufSel * 4096u;
#pragma unroll
        for (int c = 0; c < 4; ++c) {
            // each lane moves 16 B; 32 lanes x 16 B = 512 B per instruction;
            // offset:512 advances BOTH the LDS and global address (channels 128..255)
            const uint64_t ga = (uint64_t)(const void*)(fm + cols[c] + lane * 4);
            const unsigned la = bufBase + (unsigned)c * 1024u + (unsigned)lane * 16u;
            asm volatile("global_load_async_to_lds_b128 %0, %1, off"
                         :: "v"(la), "v"(ga) : "memory");
            asm volatile("global_load_async_to_lds_b128 %0, %1, off offset:512"
                         :: "v"(la), "v"(ga) : "memory");
        }
    };

    auto compute = [&](int ix, int bufSel) {
        const float x   = x1b * 63.0f + (float)ix * stepx;
        const float x0f = floorf(x);
        const float wx  = x - x0f;
        const bool ok   = iny && (x >= 0.0f) && (x <= 63.0f);
        const long long outIdx = ((long long)work * 7 + ix) * 256;
        const v4f vzero = {0.0f, 0.0f, 0.0f, 0.0f};
#pragma unroll
        for (int k = 0; k < 2; ++k) {
            const int ch = k * 128 + lane * 4;
            v4f f00 = *(const v4f*)&lds[wave][bufSel][0][ch];
            v4f f01 = *(const v4f*)&lds[wave][bufSel][1][ch];
            v4f f10 = *(const v4f*)&lds[wave][bufSel][2][ch];
            v4f f11 = *(const v4f*)&lds[wave][bufSel][3][ch];
            v4f top = f00 + (f01 - f00) * wx;
            v4f bot = f10 + (f11 - f10) * wx;
            v4f val = top + (bot - top) * wy;
            if (!ok) val = vzero;
            // streaming store: keep the L2-resident feature map hot
            __builtin_nontemporal_store(val, (v4f*)(out + outIdx + ch));
        }
    };

    // software pipeline over the 7 ix positions, double-buffered in LDS
    stage(0, 0);
#pragma unroll
    for (int ix = 0; ix < 7; ++ix) {
        if (ix < 6) {
            stage(ix + 1, (ix + 1) & 1);
            // 8 async loads outstanding for ix+1; wait until group ix landed
            asm volatile("s_wait_asynccnt 8" ::: "memory");
        } else {
            asm volatile("s_wait_asynccnt 0" ::: "memory");
        }
        compute(ix, ix & 1);
    }
}

extern "C" void kernel_launch(void* const* d_in, const int* in_sizes, int n_in,
                              void* d_out, int out_size, void* d_ws, size_t ws_size,
                              hipStream_t stream) {
    const float* fm    = (const float*)d_in[0];  // [8,64,64,256]
    const float* boxes = (const float*)d_in[1];  // [8,1000,4]
    float* out = (float*)d_out;                  // [8,1000,7,7,256]

    dim3 block(WAVES * 32);
    dim3 grid((TOTAL_WORK + WAVES - 1) / WAVES); // 14000 blocks
    roi_align_kernel<<<grid, block, 0, stream>>>(fm, boxes, out);
}